// MultiHeadAttention_33629593927773
// MI455X (gfx1250) — compile-verified
//
#include <hip/hip_runtime.h>

// ---------------------------------------------------------------------------
// MI455X (gfx1250) implementation.
//
// Roofline: feats/out are 205 MB each; total FLOPs ~1 GFLOP -> the kernel is
// HBM-bandwidth bound (~20-30 us at 23.3 TB/s). Design goals:
//   * all heavy global traffic coalesced along the contiguous H*W axis
//   * q/k projection done as a real WMMA matmul (shared weights), since
//     pev's contribution to q/k collapses to a per-(n,p) scalar correction:
//     q = feats . wq + (sum_c pev*wq + bias)
//   * branchless WMMA operand staging: B columns (wq|wk|0) come from LDS via
//     a per-lane base offset -> one ds_load_b64 per k-step, no divergence
//   * softmax/context per-pixel in registers (8x8 att), fixed-order
//     reductions for determinism (no float atomics).
// ---------------------------------------------------------------------------

typedef __attribute__((ext_vector_type(2))) float v2f;
typedef __attribute__((ext_vector_type(8))) float v8f;

static constexpr int kN  = 32;
static constexpr int kP  = 8;
static constexpr int kC  = 256;
static constexpr int kHW = 784;   // 28*28, contiguous in feats[N,P,C,H,W]
static constexpr int kNT = 49;    // 16-pixel WMMA tiles per image (49*16=784)
static constexpr int kNB = 7;     // attn blocks per image (7*112=784)

// ---------------------------------------------------------------------------
// Kernel 0: pev gather + per-(n,p) q/k corrections.
// grid = N*P blocks, 256 threads (one per c).
// ---------------------------------------------------------------------------
__global__ void prep_kernel(const int* __restrict__ deltas,
                            const float* __restrict__ pe,
                            const float* __restrict__ wq_w,
                            const float* __restrict__ wq_b,
                            const float* __restrict__ wk_w,
                            const float* __restrict__ wk_b,
                            float* __restrict__ pevbuf,
                            float* __restrict__ qcorr,
                            float* __restrict__ kcorr) {
    __shared__ float sq[256];
    __shared__ float sk[256];
    const int bid = blockIdx.x;          // n*P + p
    const int c   = threadIdx.x;
    const int d0  = deltas[bid * 2 + 0];
    const int d1  = deltas[bid * 2 + 1];
    // pe: [32, 2, 128]; pev = concat(pe[d0,0,:], pe[d1,1,:])
    const float v = (c < 128) ? pe[(d0 * 2 + 0) * 128 + c]
                              : pe[(d1 * 2 + 1) * 128 + (c - 128)];
    pevbuf[bid * kC + c] = v;
    sq[c] = v * wq_w[c];
    sk[c] = v * wk_w[c];
    __syncthreads();
    for (int s = 128; s > 0; s >>= 1) {
        if (c < s) { sq[c] += sq[c + s]; sk[c] += sk[c + s]; }
        __syncthreads();
    }
    if (c == 0) {
        qcorr[bid] = sq[0] + wq_b[0];
        kcorr[bid] = sk[0] + wk_b[0];
    }
}

// ---------------------------------------------------------------------------
// Kernel 1: q/k projection via V_WMMA_F32_16X16X4_F32.
//   D[m, n] = sum_k A[m,k] * B[k,n]
//   A: 16 pixels (M) x 4 channels (K) of raw feats -> coalesced loads
//      (A layout: lanes 0-15 K={0,1}, lanes 16-31 K={2,3}; M = lane&15)
//   B: column 0 = wq[c], column 1 = wk[c], others 0. Staged in LDS with a
//      zero block so the per-lane column select is a base offset, not a
//      branch: inner loop does one ds_load_b64 + two global_load_b32 + wmma.
//   64 chained WMMAs accumulate the full C=256 dot product (fully unrolled ->
//   immediate-offset loads the compiler can clause ahead of the WMMA chain).
// grid = (49 tiles, 8 p, 32 n), block = 32 (one wave; EXEC all ones at WMMA).
// ---------------------------------------------------------------------------
__global__ void proj_kernel(const float* __restrict__ feats,
                            const float* __restrict__ wq_w,
                            const float* __restrict__ wk_w,
                            const float* __restrict__ qcorr,
                            const float* __restrict__ kcorr,
                            float* __restrict__ qbuf,
                            float* __restrict__ kbuf) {
    __shared__ float wsh[768];           // [0,256)=wq, [256,512)=wk, [512,768)=0
    const int lane = threadIdx.x;        // 0..31

    #pragma unroll
    for (int i = 0; i < 8; ++i) {
        const int idx = lane + 32 * i;
        wsh[idx]       = wq_w[idx];
        wsh[256 + idx] = wk_w[idx];
        wsh[512 + idx] = 0.f;
    }
    __syncthreads();

    const int tile = blockIdx.x;
    const int p    = blockIdx.y;
    const int n    = blockIdx.z;
    const int m    = lane & 15;          // A row (pixel) / B column (N)
    const int khi  = lane >> 4;          // K half select (A/B VGPR layout)
    const int pixbase = tile * 16;
    const int npb  = n * kP + p;

    // Fold the per-lane K-half into the bases so the unrolled loop uses pure
    // immediate offsets.
    const float* fbase = feats + (size_t)npb * kC * kHW + pixbase + m
                               + (size_t)(2 * khi) * kHW;
    const int woff = (m == 0) ? 0 : ((m == 1) ? 256 : 512);
    const float* wptr = &wsh[woff + 2 * khi];

    v8f d = {0.f, 0.f, 0.f, 0.f, 0.f, 0.f, 0.f, 0.f};

    #pragma unroll
    for (int ks = 0; ks < 64; ++ks) {
        const int cb = ks * 4;
        v2f a, b;
        a.x = fbase[(size_t)cb * kHW];           // A vgpr0: K = 2*khi
        a.y = fbase[(size_t)cb * kHW + kHW];     // A vgpr1: K = 2*khi + 1
        b   = *(const v2f*)&wptr[cb];            // B rows K, K+1 (8B aligned)
        d = __builtin_amdgcn_wmma_f32_16x16x4_f32(
                /*neg_a=*/false, a, /*neg_b=*/false, b,
                /*c_mod=*/(short)0, d, /*reuse_a=*/false, /*reuse_b=*/false);
    }

    // C/D layout: VGPR r holds M=r (lanes 0-15, N=lane) and M=r+8 (lanes 16-31).
    // q = column N=0, k = column N=1.
    const float qc = qcorr[npb];
    const float kc = kcorr[npb];
    const int obase = npb * kHW + pixbase;
    #pragma unroll
    for (int r = 0; r < 8; ++r) {
        const float v = d[r];
        if (lane == 0)  qbuf[obase + r]     = v + qc;
        if (lane == 16) qbuf[obase + 8 + r] = v + qc;
        if (lane == 1)  kbuf[obase + r]     = v + kc;
        if (lane == 17) kbuf[obase + 8 + r] = v + kc;
    }
}

// ---------------------------------------------------------------------------
// Kernel 2: softmax + context + output. Thread = pixel (fully coalesced).
// grid = (7, 32), block = 112 (7*112 = 784 exactly, no masking).
// ---------------------------------------------------------------------------
__global__ void attn_kernel(const float* __restrict__ feats,
                            const float* __restrict__ pevbuf,
                            const float* __restrict__ qbuf,
                            const float* __restrict__ kbuf,
                            const float* __restrict__ w_o,
                            float* __restrict__ out,
                            float* __restrict__ s_part) {
    __shared__ float red[112 * 8];
    const int n   = blockIdx.y;
    const int pix = blockIdx.x * 112 + threadIdx.x;

    float qv[8], kv[8];
    #pragma unroll
    for (int p = 0; p < 8; ++p) {
        qv[p] = qbuf[(n * kP + p) * kHW + pix];
        kv[p] = kbuf[(n * kP + p) * kHW + pix];
    }

    // Row-wise stable softmax of the rank-1 score matrix q_p * k_q.
    float att[8][8];
    float ssum[8] = {0.f, 0.f, 0.f, 0.f, 0.f, 0.f, 0.f, 0.f};
    #pragma unroll
    for (int p = 0; p < 8; ++p) {
        float sc[8];
        float mx = qv[p] * kv[0];
        #pragma unroll
        for (int q = 0; q < 8; ++q) {
            sc[q] = qv[p] * kv[q];
            mx = fmaxf(mx, sc[q]);
        }
        float sum = 0.f;
        #pragma unroll
        for (int q = 0; q < 8; ++q) {
            const float e = __expf(sc[q] - mx);
            att[p][q] = e;
            sum += e;
        }
        const float inv = 1.f / sum;
        #pragma unroll
        for (int q = 0; q < 8; ++q) {
            att[p][q] *= inv;
            ssum[q] += att[p][q];
        }
    }

    // Deterministic per-block reduction of s partials.
    #pragma unroll
    for (int q = 0; q < 8; ++q) red[threadIdx.x * 8 + q] = ssum[q];
    __syncthreads();
    if (threadIdx.x < 8) {
        float acc = 0.f;
        for (int i = 0; i < 112; ++i) acc += red[i * 8 + threadIdx.x];
        s_part[(n * kNB + blockIdx.x) * 8 + threadIdx.x] = acc;
    }

    // Context + output: stream c, coalesced along pixels.
    const float* fb  = feats + (size_t)n * kP * kC * kHW + pix;
    float*       ob  = out   + (size_t)n * kP * kC * kHW + pix;
    const float* pvb = pevbuf + n * kP * kC;
    for (int c = 0; c < kC; ++c) {
        float fv[8], xv[8];
        #pragma unroll
        for (int q = 0; q < 8; ++q) {
            fv[q] = fb[(size_t)(q * kC + c) * kHW];      // feats[n,q,c,pix]
            xv[q] = fv[q] + pvb[q * kC + c];             // + pev (wave-uniform)
        }
        const float wo = w_o[c];
        #pragma unroll
        for (int p = 0; p < 8; ++p) {
            float ctx = 0.f;
            #pragma unroll
            for (int q = 0; q < 8; ++q) ctx = fmaf(att[p][q], xv[q], ctx);
            ob[(size_t)(p * kC + c) * kHW] = fmaf(wo, ctx, fv[p]);
        }
    }
}

// ---------------------------------------------------------------------------
// Kernel 3: final s reduction (fixed order -> deterministic).
// ---------------------------------------------------------------------------
__global__ void sreduce_kernel(const float* __restrict__ s_part,
                               float* __restrict__ s_out) {
    const int t = threadIdx.x;           // 0..255 -> (n, q)
    const int n = t >> 3;
    const int q = t & 7;
    float acc = 0.f;
    for (int b = 0; b < kNB; ++b) acc += s_part[(n * kNB + b) * 8 + q];
    s_out[t] = acc * (1.0f / (784.0f * 8.0f));
}

// ---------------------------------------------------------------------------
// Launch. Inputs: feats, logits(unused), deltas, pe, wq_w, wq_b, wk_w, wk_b, w_o
// Output: out [N,P,C,H,W] (51380224 f32) then s [N,P] (256 f32), concatenated.
// ---------------------------------------------------------------------------
extern "C" void kernel_launch(void* const* d_in, const int* in_sizes, int n_in,
                              void* d_out, int out_size, void* d_ws, size_t ws_size,
                              hipStream_t stream) {
    const float* feats  = (const float*)d_in[0];
    const int*   deltas = (const int*)  d_in[2];
    const float* pe     = (const float*)d_in[3];
    const float* wq_w   = (const float*)d_in[4];
    const float* wq_b   = (const float*)d_in[5];
    const float* wk_w   = (const float*)d_in[6];
    const float* wk_b   = (const float*)d_in[7];
    const float* w_o    = (const float*)d_in[8];

    float* out   = (float*)d_out;
    float* s_out = out + (size_t)kN * kP * kC * kHW;

    // Workspace layout (floats): pevbuf | qcorr | kcorr | qbuf | kbuf | s_part
    float* ws     = (float*)d_ws;
    float* pevbuf = ws;                          // N*P*C      = 65536
    float* qcorr  = pevbuf + kN * kP * kC;       // N*P        = 256
    float* kcorr  = qcorr + kN * kP;             // N*P        = 256
    float* qbuf   = kcorr + kN * kP;             // N*P*HW     = 200704
    float* kbuf   = qbuf + kN * kP * kHW;        // N*P*HW     = 200704
    float* s_part = kbuf + kN * kP * kHW;        // N*7*8      = 1792
    // total ~1.83 MB

    prep_kernel<<<dim3(kN * kP), dim3(256), 0, stream>>>(
        deltas, pe, wq_w, wq_b, wk_w, wk_b, pevbuf, qcorr, kcorr);

    proj_kernel<<<dim3(kNT, kP, kN), dim3(32), 0, stream>>>(
        feats, wq_w, wk_w, qcorr, kcorr, qbuf, kbuf);

    attn_kernel<<<dim3(kNB, kN), dim3(112), 0, stream>>>(
        feats, pevbuf, qbuf, kbuf, w_o, out, s_part);

    sreduce_kernel<<<dim3(1), dim3(256), 0, stream>>>(s_part, s_out);
}